// CinLayer_43224550867493
// MI455X (gfx1250) — compile-verified
//
#include <hip/hip_runtime.h>

// Problem constants (from reference): BS=512, TS_0=40, TS_K=128, F=64, NUM_FILTERS=128
#define BS   512
#define TS0  40
#define TSK  128   // y dimension (= WMMA K per h)
#define FDIM 64    // f dimension (= output N)
#define NF   128   // filters (= output M)

#define WSTR  132  // LDS row stride (floats) for W slice: 132 % 64 = 4 -> conflict-free ds_load_b64
#define XKSTR 68   // LDS row stride for x_k tile

typedef float v2f __attribute__((ext_vector_type(2)));
typedef float v8f __attribute__((ext_vector_type(8)));

// ---- CDNA5 async-to-LDS path (ASYNCcnt-tracked, bypasses VGPRs) --------------------
#if defined(__HIP_DEVICE_COMPILE__) && \
    __has_builtin(__builtin_amdgcn_global_load_async_to_lds_b128) && \
    __has_builtin(__builtin_amdgcn_global_load_async_to_lds_b32)  && \
    __has_builtin(__builtin_amdgcn_s_wait_asynccnt)
#define USE_ASYNC 1
#else
#define USE_ASYNC 0
#endif

#if USE_ASYNC
// Builtin signatures (from hipcc diagnostics): param0 = AS1 ("__device__") pointer to the
// transfer type (int4 for b128, int for b32), param1 = AS3 ("__shared__") pointer, then
// two immediates (inst offset, cpol).
typedef int v4i_t __attribute__((vector_size(16)));
typedef __attribute__((address_space(1))) v4i_t* g_b128_p;
typedef __attribute__((address_space(3))) v4i_t* l_b128_p;
typedef __attribute__((address_space(1))) int*   g_b32_p;
typedef __attribute__((address_space(3))) int*   l_b32_p;

__device__ __forceinline__ void async_ld_b128(const float* g, float* l) {
    __builtin_amdgcn_global_load_async_to_lds_b128((g_b128_p)g, (l_b128_p)l, 0, 0);
}
__device__ __forceinline__ void async_ld_b32(const float* g, float* l) {
    __builtin_amdgcn_global_load_async_to_lds_b32((g_b32_p)g, (l_b32_p)l, 0, 0);
}
#endif

// out[b,n,j] = bias[n] + sum_{h<40,y<128} W[n,h,y] * xk[b,y,(64h+j)/40] * x0[b,(64h+j)%40,(64h+j)/40]
__global__ __launch_bounds__(256)
void cin_wmma_f32_kernel(const float* __restrict__ x0,
                         const float* __restrict__ xk,
                         const float* __restrict__ W,
                         const float* __restrict__ bias,
                         float* __restrict__ out)
{
    extern __shared__ float smem[];
    float* sW0  = smem;                      // W slice buffer 0: NF*WSTR floats (67.6 KB)
    float* sW1  = sW0 + NF * WSTR;           // W slice buffer 1 (async double-buffer)
    float* sXK  = sW1 + NF * WSTR;           // TSK*XKSTR floats (34.8 KB)
    float* sX0T = sXK + TSK * XKSTR;         // FDIM*TS0 = 2560 floats (10.2 KB)

    const int b    = blockIdx.x;
    const int tid  = threadIdx.x;
    const int lane = tid & 31;
    const int wave = tid >> 5;               // 0..7 -> M tile n0 = 16*wave
    const int n0   = wave << 4;
    const int m    = lane & 15;              // A row / B,C column within tile
    const int kk   = (lane >> 4) << 1;       // K sub-offset: 0 (lanes 0-15), 2 (lanes 16-31)

    const float* xkb = xk + (size_t)b * (TSK * FDIM);
    const float* x0b = x0 + (size_t)b * (TS0 * FDIM);

#if USE_ASYNC
    // ---- per-h W slice stage: 4096 float4 -> 16 async b128 per thread (= 16 per wave)
    auto stage_w_async = [&](int h, float* dst) {
        const float* src = W + (size_t)h * TSK;          // row n at src + n*TS0*TSK
        #pragma unroll 4
        for (int q = tid; q < (NF * TSK) / 4; q += 256) {
            int n = q >> 5, y4 = (q & 31) << 2;
            async_ld_b128(src + (size_t)n * (TS0 * TSK) + y4, &dst[n * WSTR + y4]);
        }
    };
    // ---- initial stages: x_k (b128), x0 transposed-to-s-order (b32 scatter), W slice h=0
    for (int q = tid; q < (TSK * FDIM) / 4; q += 256) {  // 8 per thread
        int y = q >> 4, f4 = (q & 15) << 2;
        async_ld_b128(xkb + y * FDIM + f4, &sXK[y * XKSTR + f4]);
    }
    for (int idx = tid; idx < TS0 * FDIM; idx += 256) {  // 10 per thread (transpose via LDS dst addr)
        int t = idx >> 6, f = idx & 63;
        async_ld_b32(x0b + idx, &sX0T[f * TS0 + t]);
    }
    stage_w_async(0, sW0);
#else
    // ---- fallback sync staging of x_k / x0T
    for (int idx = tid; idx < TSK * FDIM; idx += 256) {
        int y = idx >> 6, f = idx & 63;
        sXK[y * XKSTR + f] = xkb[idx];
    }
    for (int idx = tid; idx < TS0 * FDIM; idx += 256) {
        int t = idx >> 6, f = idx & 63;
        sX0T[f * TS0 + t] = x0b[idx];
    }
#endif

    v8f acc[4];
    acc[0] = v8f{}; acc[1] = v8f{}; acc[2] = v8f{}; acc[3] = v8f{};

    for (int h = 0; h < TS0; ++h) {
        float* cur = (h & 1) ? sW1 : sW0;

#if USE_ASYNC
        __syncthreads();   // (a) all waves done computing h-1 -> safe to overwrite the other buffer
        if (h + 1 < TS0) {
            stage_w_async(h + 1, (h & 1) ? sW0 : sW1);   // overlap with compute(h)
            __builtin_amdgcn_s_wait_asynccnt(16);        // all but the 16 just-issued ops complete
        } else {
            __builtin_amdgcn_s_wait_asynccnt(0);
        }
        __syncthreads();   // (b) stage(h) data visible to every wave
#else
        __syncthreads();
        {   // sync stage of W[:,h,:] (128x128) with coalesced float4 loads
            const float* src = W + (size_t)h * TSK;
            for (int q = tid; q < (NF * TSK) / 4; q += 256) {
                int e = q << 2;
                int n = e >> 7, y = e & 127;
                const float4 v = *(const float4*)(src + (size_t)n * (TS0 * TSK) + y);
                float* d = &cur[n * WSTR + y];
                d[0] = v.x; d[1] = v.y; d[2] = v.z; d[3] = v.w;
            }
        }
        if (h + 1 < TS0) {
            const float* nx = W + (size_t)(h + 1) * TSK;
            int rr = tid >> 1, half = (tid & 1) << 6;
            __builtin_prefetch(nx + (size_t)rr * (TS0 * TSK) + half, 0, 0);
        }
        __syncthreads();
#endif

        // per-lane scalars for the 4 N-tiles: s = 64h + j, f = s/40, a0 = x0T[s]
        float a0[4]; int fj[4];
        #pragma unroll
        for (int jt = 0; jt < 4; ++jt) {
            int s = (h << 6) + (jt << 4) + m;
            fj[jt] = s / 40;
            a0[jt] = sX0T[s];
        }

        // ---- software-pipelined K loop: fetch fragments for step k+4 before issuing
        //      the 4 WMMAs of step k (keeps 5 LDS loads in flight under the WMMAs).
        const float* wrow = &cur[(n0 + m) * WSTR + kk];   // A row base (this lane)
        auto load_frag = [&](int k, v2f& a, v2f* braw) {
            const float* wa = wrow + k;
            a.x = wa[0]; a.y = wa[1];
            #pragma unroll
            for (int jt = 0; jt < 4; ++jt) {
                const float* xb = &sXK[(k + kk) * XKSTR + fj[jt]];
                braw[jt].x = xb[0];
                braw[jt].y = xb[XKSTR];
            }
        };

        v2f a_cur; v2f b_cur[4];
        load_frag(0, a_cur, b_cur);
        #pragma unroll 4
        for (int k0 = 0; k0 < TSK; k0 += 4) {
            v2f a_nxt; v2f b_nxt[4];
            load_frag((k0 + 4) & (TSK - 1), a_nxt, b_nxt);   // wraps on last step (harmless)
            #pragma unroll
            for (int jt = 0; jt < 4; ++jt) {
                v2f bb; bb.x = b_cur[jt].x * a0[jt]; bb.y = b_cur[jt].y * a0[jt];
                acc[jt] = __builtin_amdgcn_wmma_f32_16x16x4_f32(
                    /*neg_a=*/false, a_cur, /*neg_b=*/false, bb,
                    /*c_mod=*/(short)0, acc[jt],
                    /*reuse_a=*/false, /*reuse_b=*/false);
            }
            a_cur = a_nxt;
            #pragma unroll
            for (int jt = 0; jt < 4; ++jt) b_cur[jt] = b_nxt[jt];
        }
    }

    // ---- epilogue: C/D layout: VGPR r -> M = r + 8*(lane>=16), N = lane%16
    float* outp = out + (size_t)b * (NF * FDIM);
    const int rowbase = n0 + ((lane >> 4) << 3);
    #pragma unroll
    for (int r = 0; r < 8; ++r) {
        int n = rowbase + r;
        float bv = bias[n];
        outp[n * FDIM +  0 + m] = acc[0][r] + bv;
        outp[n * FDIM + 16 + m] = acc[1][r] + bv;
        outp[n * FDIM + 32 + m] = acc[2][r] + bv;
        outp[n * FDIM + 48 + m] = acc[3][r] + bv;
    }
}

extern "C" void kernel_launch(void* const* d_in, const int* in_sizes, int n_in,
                              void* d_out, int out_size, void* d_ws, size_t ws_size,
                              hipStream_t stream) {
    const float* x0   = (const float*)d_in[0];   // (512, 40, 64)
    const float* xk   = (const float*)d_in[1];   // (512, 128, 64)
    const float* W    = (const float*)d_in[2];   // (128, 1, 40, 128)
    const float* bias = (const float*)d_in[3];   // (128,)
    float* out = (float*)d_out;                  // (512, 128, 64)

    // Always size LDS for the double-buffered layout (device picks sync path into buffer h&1 otherwise)
    const size_t smem_bytes = (size_t)(2 * NF * WSTR + TSK * XKSTR + FDIM * TS0) * sizeof(float);
    dim3 grid(BS), block(256);
    cin_wmma_f32_kernel<<<grid, block, smem_bytes, stream>>>(x0, xk, W, bias, out);
}